// QubitE_36953898615101
// MI455X (gfx1250) — compile-verified
//
#include <hip/hip_runtime.h>

#define N_ENT 50000
#define N_REL 500
#define DIM   256
#define BATCH 512
#define BN_EPS 1e-5f

typedef float v2f __attribute__((ext_vector_type(2)));
typedef float v8f __attribute__((ext_vector_type(8)));

// ---- workspace layout (float offsets) ----
#define WS_ESUM   0        // 4*256 sums of qnorm(E) over all entities
#define WS_ESQ    1024     // 4*256 sums of squares
#define WS_HSUM   2048     // 4*256 sums of qnorm(E[h_idx]) over batch
#define WS_HSQ    3072
#define WS_ESCALE 4096     // folded BN scale for E table
#define WS_ESHIFT 5120     // folded BN shift for E table
#define WS_T      6144     // t matrix [4][512][256]
// total: 6144 + 524288 floats  ~= 2.1 MB

// ---------------------------------------------------------------------------
__global__ void k_zero_stats(float* __restrict__ ws) {
  int i = blockIdx.x * blockDim.x + threadIdx.x;
  if (i < 4096) ws[i] = 0.0f;
}

// Stats of qnorm(E) over all 50000 entities. 250 blocks * 200 rows, thread = dim.
__global__ void k_estats(const float* __restrict__ E, float* __restrict__ ws) {
  const int d = threadIdx.x;                    // 0..255 == DIM
  const int row0 = blockIdx.x * 200;
  float s0 = 0.f, s1 = 0.f, s2 = 0.f, s3 = 0.f;
  float q0s = 0.f, q1s = 0.f, q2s = 0.f, q3s = 0.f;
  for (int r = 0; r < 200; ++r) {
    const int n = row0 + r;
    const float e0 = E[(size_t)(0 * N_ENT + n) * DIM + d];
    const float e1 = E[(size_t)(1 * N_ENT + n) * DIM + d];
    const float e2 = E[(size_t)(2 * N_ENT + n) * DIM + d];
    const float e3 = E[(size_t)(3 * N_ENT + n) * DIM + d];
    const float inv = rsqrtf(e0 * e0 + e1 * e1 + e2 * e2 + e3 * e3);
    const float q0 = e0 * inv, q1 = e1 * inv, q2 = e2 * inv, q3 = e3 * inv;
    s0 += q0;  s1 += q1;  s2 += q2;  s3 += q3;
    q0s += q0 * q0; q1s += q1 * q1; q2s += q2 * q2; q3s += q3 * q3;
  }
  atomicAdd(&ws[WS_ESUM + 0 * DIM + d], s0);
  atomicAdd(&ws[WS_ESUM + 1 * DIM + d], s1);
  atomicAdd(&ws[WS_ESUM + 2 * DIM + d], s2);
  atomicAdd(&ws[WS_ESUM + 3 * DIM + d], s3);
  atomicAdd(&ws[WS_ESQ + 0 * DIM + d], q0s);
  atomicAdd(&ws[WS_ESQ + 1 * DIM + d], q1s);
  atomicAdd(&ws[WS_ESQ + 2 * DIM + d], q2s);
  atomicAdd(&ws[WS_ESQ + 3 * DIM + d], q3s);
}

// Stats of qnorm(E[h_idx]) over the 512-sample batch. 32 blocks * 16 rows.
__global__ void k_hstats(const float* __restrict__ E, const int* __restrict__ h_idx,
                         float* __restrict__ ws) {
  const int d = threadIdx.x;
  const int b0 = blockIdx.x * 16;
  float s0 = 0.f, s1 = 0.f, s2 = 0.f, s3 = 0.f;
  float q0s = 0.f, q1s = 0.f, q2s = 0.f, q3s = 0.f;
  for (int r = 0; r < 16; ++r) {
    const int n = h_idx[b0 + r];
    const float e0 = E[(size_t)(0 * N_ENT + n) * DIM + d];
    const float e1 = E[(size_t)(1 * N_ENT + n) * DIM + d];
    const float e2 = E[(size_t)(2 * N_ENT + n) * DIM + d];
    const float e3 = E[(size_t)(3 * N_ENT + n) * DIM + d];
    const float inv = rsqrtf(e0 * e0 + e1 * e1 + e2 * e2 + e3 * e3);
    const float q0 = e0 * inv, q1 = e1 * inv, q2 = e2 * inv, q3 = e3 * inv;
    s0 += q0;  s1 += q1;  s2 += q2;  s3 += q3;
    q0s += q0 * q0; q1s += q1 * q1; q2s += q2 * q2; q3s += q3 * q3;
  }
  atomicAdd(&ws[WS_HSUM + 0 * DIM + d], s0);
  atomicAdd(&ws[WS_HSUM + 1 * DIM + d], s1);
  atomicAdd(&ws[WS_HSUM + 2 * DIM + d], s2);
  atomicAdd(&ws[WS_HSUM + 3 * DIM + d], s3);
  atomicAdd(&ws[WS_HSQ + 0 * DIM + d], q0s);
  atomicAdd(&ws[WS_HSQ + 1 * DIM + d], q1s);
  atomicAdd(&ws[WS_HSQ + 2 * DIM + d], q2s);
  atomicAdd(&ws[WS_HSQ + 3 * DIM + d], q3s);
}

// Fold BN into Ehat = q * scale + shift.
__global__ void k_finalize_escale(const float* __restrict__ gamma,
                                  const float* __restrict__ beta,
                                  float* __restrict__ ws) {
  const int i = blockIdx.x * blockDim.x + threadIdx.x;   // c*DIM + d
  if (i >= 4 * DIM) return;
  const float mean = ws[WS_ESUM + i] * (1.0f / (float)N_ENT);
  const float var  = ws[WS_ESQ + i] * (1.0f / (float)N_ENT) - mean * mean;
  const float istd = rsqrtf(var + BN_EPS);
  const float g = gamma[i];
  ws[WS_ESCALE + i] = g * istd;
  ws[WS_ESHIFT + i] = beta[i] - mean * g * istd;
}

// Build t[4][512][256]: gather h & r, qnorm, batch-BN on h, qubit multiply.
__global__ void k_compute_t(const float* __restrict__ E, const float* __restrict__ R,
                            const float* __restrict__ psi,
                            const float* __restrict__ gamma, const float* __restrict__ beta,
                            const int* __restrict__ h_idx, const int* __restrict__ r_idx,
                            float* __restrict__ ws) {
  const int b = blockIdx.x;        // 0..511
  const int d = threadIdx.x;       // 0..255
  const int hn = h_idx[b];
  const int rn = r_idx[b];

  // h path: qnorm then batch BN (stats over 512 samples)
  float q[4], hh[4];
  {
    const float e0 = E[(size_t)(0 * N_ENT + hn) * DIM + d];
    const float e1 = E[(size_t)(1 * N_ENT + hn) * DIM + d];
    const float e2 = E[(size_t)(2 * N_ENT + hn) * DIM + d];
    const float e3 = E[(size_t)(3 * N_ENT + hn) * DIM + d];
    const float inv = rsqrtf(e0 * e0 + e1 * e1 + e2 * e2 + e3 * e3);
    q[0] = e0 * inv; q[1] = e1 * inv; q[2] = e2 * inv; q[3] = e3 * inv;
  }
#pragma unroll
  for (int c = 0; c < 4; ++c) {
    const float mean = ws[WS_HSUM + c * DIM + d] * (1.0f / (float)BATCH);
    const float var  = ws[WS_HSQ + c * DIM + d] * (1.0f / (float)BATCH) - mean * mean;
    const float istd = rsqrtf(var + BN_EPS);
    hh[c] = (q[c] - mean) * istd * gamma[c * DIM + d] + beta[c * DIM + d];
  }

  // r path: qnorm only
  float rr[4];
  {
    const float r0 = R[(size_t)(0 * N_REL + rn) * DIM + d];
    const float r1 = R[(size_t)(1 * N_REL + rn) * DIM + d];
    const float r2 = R[(size_t)(2 * N_REL + rn) * DIM + d];
    const float r3 = R[(size_t)(3 * N_REL + rn) * DIM + d];
    const float inv = rsqrtf(r0 * r0 + r1 * r1 + r2 * r2 + r3 * r3);
    rr[0] = r0 * inv; rr[1] = r1 * inv; rr[2] = r2 * inv; rr[3] = r3 * inv;
  }

  const float p = psi[(size_t)rn * DIM + d];
  const float cz = cosf(p), sz = sinf(p);

  // QubitMult
  const float m1re = hh[0] * rr[0] - hh[1] * rr[1];
  const float m1im = hh[0] * rr[1] + hh[1] * rr[0];
  const float X = hh[2] * rr[2] + hh[3] * rr[3];       // h_b * conj(r_b)
  const float Y = hh[3] * rr[2] - hh[2] * rr[3];
  const float m2re = cz * X - sz * Y;
  const float m2im = cz * Y + sz * X;
  const float m3re = hh[0] * rr[2] - hh[1] * rr[3];
  const float m3im = hh[0] * rr[3] + hh[1] * rr[2];
  const float U = hh[2] * rr[0] + hh[3] * rr[1];       // h_b * conj(r_a)
  const float V = hh[3] * rr[0] - hh[2] * rr[1];
  const float m4re = cz * U - sz * V;
  const float m4im = cz * V + sz * U;

  float t0 = m1re - m2re;
  float t1 = m1im - m2im;
  float t2 = m3re + m4re;
  float t3 = m3im + m4im;

  float* T = ws + WS_T;
  T[((size_t)(0 * BATCH) + b) * DIM + d] = t0;
  T[((size_t)(1 * BATCH) + b) * DIM + d] = t1;
  T[((size_t)(2 * BATCH) + b) * DIM + d] = t2;
  T[((size_t)(3 * BATCH) + b) * DIM + d] = t3;
}

// ---------------------------------------------------------------------------
// GEMM + bias + sigmoid: out[c][b][n] = sigmoid(sum_d t[c][b][d]*Ehat[c][n][d] + bias[c][n])
// Macro tile: M=64 (batch) x N=64 (entities) x all 4 components per 256-thread block.
// K staged in chunks of 16; E is normalized+BN'd during global->LDS staging.
// f32 WMMA 16x16x4; fragment layouts per CDNA5 ISA:
//   A: lane l (m=l&15, h=l>>4), vgpr v -> K = 2h+v
//   B: lane l (n=l&15, h=l>>4), vgpr v -> K = 2h+v
//   C/D: lane l (n=l&15), vgpr v -> M = v + 8*(l>>4)
#define NT 64
#define MT 64
#define KC 16
#define LDP 18   // padded row stride (floats), even for float2 alignment

__global__ __launch_bounds__(256) void k_gemm(const float* __restrict__ E,
                                              const float* __restrict__ biases,
                                              const float* __restrict__ ws,
                                              float* __restrict__ out) {
  __shared__ float sA[4][MT][LDP];
  __shared__ float sB[4][NT][LDP];
  __shared__ float sScale[4 * DIM];
  __shared__ float sShift[4 * DIM];

  const int tid = threadIdx.x;
  const int nbase = blockIdx.x * NT;
  const int mbase = blockIdx.y * MT;
  const float* __restrict__ T = ws + WS_T;

  for (int i = tid; i < 4 * DIM; i += 256) {
    sScale[i] = ws[WS_ESCALE + i];
    sShift[i] = ws[WS_ESHIFT + i];
  }
  __syncthreads();

  const int lane = tid & 31;
  const int wid = tid >> 5;         // 0..7
  const int l15 = lane & 15;
  const int hh = lane >> 4;         // 0/1
  const int mtile = (wid & 3) * 16;             // wave's m sub-tile
  const int ntile0 = ((wid >> 2) * 2) * 16;     // wave's first n sub-tile (handles 2)

  v8f acc[2][4];
#pragma unroll
  for (int p = 0; p < 2; ++p)
#pragma unroll
    for (int c = 0; c < 4; ++c)
      acc[p][c] = (v8f){0.f, 0.f, 0.f, 0.f, 0.f, 0.f, 0.f, 0.f};

  const int skk = tid & 15;     // staging k within chunk
  const int sr0 = tid >> 4;     // staging row 0..15

  for (int chunk = 0; chunk < DIM / KC; ++chunk) {
    const int k0 = chunk * KC;
    const int k = k0 + skk;
    // ---- stage B: normalize + BN raw E on the fly ----
#pragma unroll
    for (int p = 0; p < 4; ++p) {
      const int nn = sr0 + p * 16;            // 0..63
      const int n = nbase + nn;
      const int nc = (n < N_ENT) ? n : (N_ENT - 1);
      const float e0 = E[(size_t)(0 * N_ENT + nc) * DIM + k];
      const float e1 = E[(size_t)(1 * N_ENT + nc) * DIM + k];
      const float e2 = E[(size_t)(2 * N_ENT + nc) * DIM + k];
      const float e3 = E[(size_t)(3 * N_ENT + nc) * DIM + k];
      if (k0 + KC < DIM)  // prefetch next K-chunk (global_prefetch_b8)
        __builtin_prefetch(&E[(size_t)(0 * N_ENT + nc) * DIM + k + KC], 0, 0);
      const float inv = rsqrtf(e0 * e0 + e1 * e1 + e2 * e2 + e3 * e3);
      sB[0][nn][skk] = fmaf(e0 * inv, sScale[0 * DIM + k], sShift[0 * DIM + k]);
      sB[1][nn][skk] = fmaf(e1 * inv, sScale[1 * DIM + k], sShift[1 * DIM + k]);
      sB[2][nn][skk] = fmaf(e2 * inv, sScale[2 * DIM + k], sShift[2 * DIM + k]);
      sB[3][nn][skk] = fmaf(e3 * inv, sScale[3 * DIM + k], sShift[3 * DIM + k]);
    }
    // ---- stage A: t tile (plain copy) ----
#pragma unroll
    for (int p = 0; p < 4; ++p) {
      const int mm = sr0 + p * 16;            // 0..63
#pragma unroll
      for (int c = 0; c < 4; ++c)
        sA[c][mm][skk] = T[((size_t)(c * BATCH) + (mbase + mm)) * DIM + k];
    }
    __syncthreads();

    // ---- compute: 4 K-steps of 4, 4 components, 2 n-tiles ----
#pragma unroll
    for (int s = 0; s < KC / 4; ++s) {
      const int koff = 4 * s + 2 * hh;
#pragma unroll
      for (int c = 0; c < 4; ++c) {
        const v2f a  = *(const v2f*)&sA[c][mtile + l15][koff];
        const v2f b0 = *(const v2f*)&sB[c][ntile0 + l15][koff];
        const v2f b1 = *(const v2f*)&sB[c][ntile0 + 16 + l15][koff];
        acc[0][c] = __builtin_amdgcn_wmma_f32_16x16x4_f32(
            false, a, false, b0, (short)0, acc[0][c], false, false);
        acc[1][c] = __builtin_amdgcn_wmma_f32_16x16x4_f32(
            false, a, false, b1, (short)0, acc[1][c], false, false);
      }
    }
    __syncthreads();
  }

  // ---- epilogue: bias + sigmoid + store ----
#pragma unroll
  for (int p = 0; p < 2; ++p) {
    const int nb = nbase + ntile0 + p * 16 + l15;
    if (nb < N_ENT) {
#pragma unroll
      for (int c = 0; c < 4; ++c) {
        const float bias = biases[(size_t)c * N_ENT + nb];
#pragma unroll
        for (int v = 0; v < 8; ++v) {
          const int m = mbase + mtile + v + 8 * hh;
          const float x = acc[p][c][v] + bias;
          const float y = 1.0f / (1.0f + __expf(-x));
          out[((size_t)(c * BATCH) + m) * N_ENT + nb] = y;
        }
      }
    }
  }
}

// ---------------------------------------------------------------------------
extern "C" void kernel_launch(void* const* d_in, const int* in_sizes, int n_in,
                              void* d_out, int out_size, void* d_ws, size_t ws_size,
                              hipStream_t stream) {
  (void)in_sizes; (void)n_in; (void)out_size; (void)ws_size;
  const float* E      = (const float*)d_in[0];
  const float* R      = (const float*)d_in[1];
  const float* psi    = (const float*)d_in[2];
  const float* gamma  = (const float*)d_in[3];
  const float* beta   = (const float*)d_in[4];
  const float* biases = (const float*)d_in[5];
  const int*   h_idx  = (const int*)d_in[6];
  const int*   r_idx  = (const int*)d_in[7];
  float* out = (float*)d_out;
  float* ws  = (float*)d_ws;

  hipLaunchKernelGGL(k_zero_stats, dim3(16), dim3(256), 0, stream, ws);
  hipLaunchKernelGGL(k_estats, dim3(250), dim3(256), 0, stream, E, ws);
  hipLaunchKernelGGL(k_hstats, dim3(32), dim3(256), 0, stream, E, h_idx, ws);
  hipLaunchKernelGGL(k_finalize_escale, dim3(4), dim3(256), 0, stream, gamma, beta, ws);
  hipLaunchKernelGGL(k_compute_t, dim3(BATCH), dim3(256), 0, stream,
                     E, R, psi, gamma, beta, h_idx, r_idx, ws);
  dim3 g((N_ENT + NT - 1) / NT, BATCH / MT);
  hipLaunchKernelGGL(k_gemm, g, dim3(256), 0, stream, E, biases, ws, out);
}